// MolecularGraphEncoder_75282186764806
// MI455X (gfx1250) — compile-verified
//
#include <hip/hip_runtime.h>
#include <math.h>

// ---------------------------------------------------------------------------
// MolecularGraphEncoder for MI455X (gfx1250, wave32, WMMA).
// Sizes fixed by the reference.
// ---------------------------------------------------------------------------
namespace {
constexpr int kN  = 50000;    // atoms
constexpr int kD  = 128;      // d_model
constexpr int kH  = 8;        // heads
constexpr int kHD = 16;       // head dim
constexpr int kL  = 4;        // layers
constexpr int kE  = 400000;   // undirected edges (2E directed)
}

typedef float v2f __attribute__((ext_vector_type(2)));
typedef float v8f __attribute__((ext_vector_type(8)));
typedef int   v4i __attribute__((ext_vector_type(4)));

#if __has_builtin(__builtin_amdgcn_global_load_async_to_lds_b128)
#define HAS_ASYNC_LDS 1
typedef __attribute__((address_space(1))) v4i* as1_v4i;
typedef __attribute__((address_space(3))) v4i* as3_v4i;
#else
#define HAS_ASYNC_LDS 0
#endif

// Order-preserving f32 <-> u32 encoding so segment-max can use atomicMax(u32).
__device__ __forceinline__ unsigned fenc(float f) {
  unsigned u = __float_as_uint(f);
  return (u & 0x80000000u) ? ~u : (u | 0x80000000u);
}
__device__ __forceinline__ float fdec(unsigned u) {
  unsigned v = (u & 0x80000000u) ? (u & 0x7fffffffu) : ~u;
  return __uint_as_float(v);
}

// ---------------------------------------------------------------------------
// init / degree / edge-feature segment sum (runs once)
// ---------------------------------------------------------------------------
__global__ void init_u32_kernel(unsigned* __restrict__ p, unsigned val, int n) {
  int i = blockIdx.x * blockDim.x + threadIdx.x;
  if (i < n) p[i] = val;
}

// thread = (directed edge, 4-float chunk); 2E*32 threads.
__global__ void deg_efsum_kernel(const int* __restrict__ ei,
                                 const float* __restrict__ ef,
                                 float* __restrict__ deg,
                                 float* __restrict__ efsum) {
  int gid = blockIdx.x * blockDim.x + threadIdx.x;
  if (gid >= 2 * kE * 32) return;
  int c = gid & 31;
  int e = gid >> 5;                       // directed edge id in [0, 2E)
  int t = ei[e];                          // receiver: concat(src, dst)[e] == ei[e]
  int ue = (e < kE) ? e : (e - kE);       // undirected edge id
  if (c == 0) atomicAdd(&deg[t], 1.0f);
  float4 val = ((const float4*)ef)[(size_t)ue * 32 + c];
  float* dp = efsum + (size_t)t * kD + c * 4;
  atomicAdd(dp + 0, val.x);
  atomicAdd(dp + 1, val.y);
  atomicAdd(dp + 2, val.z);
  atomicAdd(dp + 3, val.w);
}

// ---------------------------------------------------------------------------
// GEMM: Y[M x 128] = X[M x 128] @ W^T + bias, W row-major [128 x 128].
// fp32 WMMA 16x16x4. W staged into LDS once per workgroup via the gfx1250
// async global->LDS path (ASYNCcnt); B fragments then come from LDS.
// One wave = 16-row x 128-col strip (8 C tiles).
// ---------------------------------------------------------------------------
__global__ __launch_bounds__(128) void gemm128_wmma_kernel(
    const float* __restrict__ X, const float* __restrict__ W,
    const float* __restrict__ bias, float* __restrict__ Y, int M) {
  __shared__ float Wlds[kD * kD];                 // 64 KB of the WGP's 320 KB

  // ---- stage W: 16384 floats, 128 threads x 32 x float4 ----
  {
    const int tid = threadIdx.x;
#if HAS_ASYNC_LDS
#pragma unroll
    for (int i = 0; i < 32; ++i) {
      const int idx = (i * 128 + tid) * 4;        // float index, 16B aligned
      __builtin_amdgcn_global_load_async_to_lds_b128(
          (as1_v4i)(W + idx), (as3_v4i)(Wlds + idx), 0, 0);
    }
#if __has_builtin(__builtin_amdgcn_s_wait_asynccnt)
    __builtin_amdgcn_s_wait_asynccnt(0);
#else
    asm volatile("s_wait_asynccnt 0x0" ::: "memory");
#endif
#else
#pragma unroll
    for (int i = 0; i < 32; ++i) {
      const int idx = i * 128 + tid;
      ((float4*)Wlds)[idx] = ((const float4*)W)[idx];
    }
#endif
    __syncthreads();
  }

  const int wave = threadIdx.x >> 5;
  const int lane = threadIdx.x & 31;
  const int row0 = blockIdx.x * 64 + wave * 16;
  if (row0 >= M) return;                       // uniform per wave (EXEC stays full)

  const int m  = lane & 15;                    // A: M index / B,C: N index
  const int kh = lane >> 4;                    // half-select: K offset 0 or 2

  const float2* Arow = (const float2*)(X + (size_t)(row0 + m) * kD);
  const float2* Brow[8];
#pragma unroll
  for (int t = 0; t < 8; ++t)
    Brow[t] = (const float2*)(Wlds + (size_t)(t * 16 + m) * kD);  // B[k][n] = W[n][k]

  v8f acc[8];
#pragma unroll
  for (int t = 0; t < 8; ++t)
#pragma unroll
    for (int r = 0; r < 8; ++r) acc[t][r] = 0.0f;

  // K = 128, 4 per step -> 32 steps; float2 index = k0/2 (+kh)
#pragma unroll 4
  for (int k2 = 0; k2 < 64; k2 += 2) {
    float2 af = Arow[k2 + kh];
    v2f a; a.x = af.x; a.y = af.y;             // A lane: (M=m, K=k0+2*kh, +1)
#pragma unroll
    for (int t = 0; t < 8; ++t) {
      float2 bf = Brow[t][k2 + kh];
      v2f b; b.x = bf.x; b.y = bf.y;           // B lane: (K=k0+2*kh.., N=m)
      acc[t] = __builtin_amdgcn_wmma_f32_16x16x4_f32(
          false, a, false, b, (short)0, acc[t], false, false);
    }
  }

  const int mb = kh * 8;                       // C/D: lanes16-31 hold M=8..15
#pragma unroll
  for (int t = 0; t < 8; ++t) {
    const int col = t * 16 + m;
    float bb = bias ? bias[col] : 0.0f;
#pragma unroll
    for (int r = 0; r < 8; ++r)
      Y[(size_t)(row0 + mb + r) * kD + col] = acc[t][r] + bb;
  }
}

// ---------------------------------------------------------------------------
// Per-node attention bias from edge projection (via linearity of We):
// bias[n,h] = sum_{d in head h}(bsum[n,d] + deg[n]*be[d]) / (16*max(deg,1))
// ---------------------------------------------------------------------------
__global__ void bias_kernel(const float* __restrict__ bsum,
                            const float* __restrict__ be,
                            const float* __restrict__ deg,
                            float* __restrict__ bias) {
  int gid = blockIdx.x * blockDim.x + threadIdx.x;
  if (gid >= kN * kH) return;
  int n = gid >> 3, h = gid & 7;
  float dg = deg[n];
  const float* bp  = bsum + (size_t)n * kD + h * kHD;
  const float* bep = be + h * kHD;
  float s = 0.f;
#pragma unroll
  for (int i = 0; i < kHD; ++i) s += bp[i] + dg * bep[i];
  bias[gid] = s / (16.0f * fmaxf(dg, 1.0f));
}

// ---------------------------------------------------------------------------
// Phase A: per-(directed edge, head) score + segment max (encoded atomicMax)
// ---------------------------------------------------------------------------
__global__ void edge_scores_kernel(const int* __restrict__ ei,
                                   const float* __restrict__ q,
                                   const float* __restrict__ k,
                                   const float* __restrict__ bias,
                                   float* __restrict__ scores,
                                   unsigned* __restrict__ menc) {
  int gid = blockIdx.x * blockDim.x + threadIdx.x;
  if (gid >= 2 * kE * kH) return;
  int h = gid & 7;
  int e = gid >> 3;                                  // directed edge
  int t = ei[e];
  int s = (e < kE) ? ei[e + kE] : ei[e - kE];
  const float4* qp = (const float4*)(q + (size_t)t * kD + h * kHD);
  const float4* kp = (const float4*)(k + (size_t)s * kD + h * kHD);
  float dot = 0.f;
#pragma unroll
  for (int i = 0; i < 4; ++i) {
    float4 a = qp[i], b = kp[i];
    dot += a.x * b.x + a.y * b.y + a.z * b.z + a.w * b.w;
  }
  float sc = dot * 0.25f + bias[t * kH + h];          // inv_scale = 1/sqrt(16)
  scores[(size_t)e * kH + h] = sc;
  atomicMax(&menc[t * kH + h], fenc(sc));
}

// ---------------------------------------------------------------------------
// Phase B: p = exp(score - m); den += p; agg += p * v[s]  (unnormalized)
// ---------------------------------------------------------------------------
__global__ void edge_soft_agg_kernel(const int* __restrict__ ei,
                                     const float* __restrict__ scores,
                                     const unsigned* __restrict__ menc,
                                     const float* __restrict__ v,
                                     float* __restrict__ den,
                                     float* __restrict__ agg) {
  int gid = blockIdx.x * blockDim.x + threadIdx.x;
  if (gid >= 2 * kE * kH) return;
  int h = gid & 7;
  int e = gid >> 3;
  int t = ei[e];
  int s = (e < kE) ? ei[e + kE] : ei[e - kE];
  float m = fdec(menc[t * kH + h]);
  if (!isfinite(m)) m = 0.f;
  float p = expf(scores[(size_t)e * kH + h] - m);
  atomicAdd(&den[t * kH + h], p);
  const float* vp = v + (size_t)s * kD + h * kHD;
  float* ap = agg + (size_t)t * kD + h * kHD;
#pragma unroll
  for (int i = 0; i < kHD; ++i) atomicAdd(&ap[i], p * vp[i]);
}

__global__ void normalize_agg_kernel(float* __restrict__ agg,
                                     const float* __restrict__ den) {
  int gid = blockIdx.x * blockDim.x + threadIdx.x;
  if (gid >= kN * kD) return;
  int n = gid >> 7, d = gid & 127;
  agg[gid] /= fmaxf(den[n * kH + (d >> 4)], 1e-30f);
}

// ---------------------------------------------------------------------------
// x = LayerNorm( where(deg>0, yo, x) + x );  one wave32 per row, in place.
// ---------------------------------------------------------------------------
__global__ __launch_bounds__(256) void resid_ln_kernel(
    const float* __restrict__ yo, float* __restrict__ x,
    const float* __restrict__ deg, const float* __restrict__ g,
    const float* __restrict__ b) {
  int wid  = (blockIdx.x * blockDim.x + threadIdx.x) >> 5;
  int lane = threadIdx.x & 31;
  if (wid >= kN) return;
  const float4* y4 = (const float4*)(yo + (size_t)wid * kD);
  float4* x4 = (float4*)(x + (size_t)wid * kD);
  float4 yv = y4[lane];
  float4 xv = x4[lane];
  if (deg[wid] <= 0.f) yv = xv;                       // isolated-node passthrough
  float4 sv;
  sv.x = yv.x + xv.x; sv.y = yv.y + xv.y; sv.z = yv.z + xv.z; sv.w = yv.w + xv.w;
  float sum = sv.x + sv.y + sv.z + sv.w;
#pragma unroll
  for (int off = 16; off >= 1; off >>= 1) sum += __shfl_xor(sum, off, 32);
  float mu = sum * (1.0f / 128.0f);
  float4 dv;
  dv.x = sv.x - mu; dv.y = sv.y - mu; dv.z = sv.z - mu; dv.w = sv.w - mu;
  float sq = dv.x * dv.x + dv.y * dv.y + dv.z * dv.z + dv.w * dv.w;
#pragma unroll
  for (int off = 16; off >= 1; off >>= 1) sq += __shfl_xor(sq, off, 32);
  float r = rsqrtf(sq * (1.0f / 128.0f) + 1e-5f);
  float4 gv = ((const float4*)g)[lane];
  float4 bv = ((const float4*)b)[lane];
  float4 ov;
  ov.x = dv.x * r * gv.x + bv.x;
  ov.y = dv.y * r * gv.y + bv.y;
  ov.z = dv.z * r * gv.z + bv.z;
  ov.w = dv.w * r * gv.w + bv.w;
  x4[lane] = ov;
}

// ---------------------------------------------------------------------------
// pooled mean + 2-layer MLP head
// ---------------------------------------------------------------------------
__global__ __launch_bounds__(128) void pool_reduce_kernel(
    const float* __restrict__ x, float* __restrict__ pool) {
  int d = threadIdx.x;
  float s = 0.f;
  for (int n = blockIdx.x; n < kN; n += gridDim.x) s += x[(size_t)n * kD + d];
  atomicAdd(&pool[d], s);
}

__global__ __launch_bounds__(128) void mlp_kernel(
    const float* __restrict__ pool, const float* __restrict__ W1,
    const float* __restrict__ b1, const float* __restrict__ W2,
    const float* __restrict__ b2, float* __restrict__ out) {
  __shared__ float pm[kD];
  __shared__ float hh[kD];
  int tid = threadIdx.x;
  pm[tid] = pool[tid] * (1.0f / (float)kN);
  __syncthreads();
  float a = b1[tid];
  for (int i = 0; i < kD; ++i) a += pm[i] * W1[tid * kD + i];
  hh[tid] = fmaxf(a, 0.f);
  __syncthreads();
  float o = b2[tid];
  for (int i = 0; i < kD; ++i) o += hh[i] * W2[tid * kD + i];
  out[tid] = o;
}

// ---------------------------------------------------------------------------
// host launcher
// ---------------------------------------------------------------------------
static inline int divUp(int a, int b) { return (a + b - 1) / b; }

extern "C" void kernel_launch(void* const* d_in, const int* in_sizes, int n_in,
                              void* d_out, int out_size, void* d_ws, size_t ws_size,
                              hipStream_t stream) {
  const float* nodef = (const float*)d_in[0];
  const int*   ei    = (const int*)d_in[1];
  const float* ef    = (const float*)d_in[2];

  // params in setup_inputs() dict order (fallback: one concatenated blob)
  const float* P[16];
  if (n_in >= 19) {
    for (int i = 0; i < 16; ++i) P[i] = (const float*)d_in[3 + i];
  } else {
    static const size_t sz[16] = {65536, 512, 65536, 512, 65536, 512, 65536, 512,
                                  65536, 512, 512, 512, 16384, 128, 16384, 128};
    const float* base = (const float*)d_in[3];
    size_t off = 0;
    for (int i = 0; i < 16; ++i) { P[i] = base + off; off += sz[i]; }
  }
  const float *Wq = P[0], *bq = P[1], *Wk = P[2], *bk = P[3];
  const float *Wv = P[4], *bv = P[5], *We = P[6], *be = P[7];
  const float *Wo = P[8], *bo = P[9], *lng = P[10], *lnb = P[11];
  const float *pW1 = P[12], *pb1 = P[13], *pW2 = P[14], *pb2 = P[15];

  // workspace layout
  float* ws = (float*)d_ws;
  size_t off = 0;
  float* q     = ws + off; off += (size_t)kN * kD;
  float* kk    = ws + off; off += (size_t)kN * kD;
  float* vv    = ws + off; off += (size_t)kN * kD;
  float* efsum = ws + off; off += (size_t)kN * kD;
  float* agg   = ws + off; off += (size_t)kN * kD;   // doubles as bsum
  float* yo    = ws + off; off += (size_t)kN * kD;
  float* deg   = ws + off; off += (size_t)kN;
  float* biasb = ws + off; off += (size_t)kN * kH;
  float* den   = ws + off; off += (size_t)kN * kH;
  unsigned* menc = (unsigned*)(ws + off); off += (size_t)kN * kH;
  float* scores = ws + off; off += (size_t)2 * kE * kH;
  float* pool   = ws + off; off += kD;

  float* x   = (float*)d_out;            // node states live in d_out
  float* emb = x + (size_t)kN * kD;      // graph embedding tail

  // x <- node_features ; deg/ef_sum once (linearity of We projection)
  (void)hipMemcpyAsync(x, nodef, (size_t)kN * kD * sizeof(float),
                       hipMemcpyDeviceToDevice, stream);
  (void)hipMemsetAsync(deg, 0, (size_t)kN * sizeof(float), stream);
  (void)hipMemsetAsync(efsum, 0, (size_t)kN * kD * sizeof(float), stream);
  deg_efsum_kernel<<<divUp(2 * kE * 32, 256), 256, 0, stream>>>(ei, ef, deg, efsum);

  const int gemmGrid = divUp(kN, 64);
  const int edgeGrid = divUp(2 * kE * kH, 256);

  for (int l = 0; l < kL; ++l) {
    const size_t wl = (size_t)l * kD * kD, bl = (size_t)l * kD;
    gemm128_wmma_kernel<<<gemmGrid, 128, 0, stream>>>(x, Wq + wl, bq + bl, q, kN);
    gemm128_wmma_kernel<<<gemmGrid, 128, 0, stream>>>(x, Wk + wl, bk + bl, kk, kN);
    gemm128_wmma_kernel<<<gemmGrid, 128, 0, stream>>>(x, Wv + wl, bv + bl, vv, kN);
    gemm128_wmma_kernel<<<gemmGrid, 128, 0, stream>>>(efsum, We + wl, nullptr, agg, kN);
    bias_kernel<<<divUp(kN * kH, 256), 256, 0, stream>>>(agg, be + bl, deg, biasb);

    init_u32_kernel<<<divUp(kN * kH, 256), 256, 0, stream>>>(
        menc, 0x007fffffu /* fenc(-inf) */, kN * kH);
    (void)hipMemsetAsync(den, 0, (size_t)kN * kH * sizeof(float), stream);
    (void)hipMemsetAsync(agg, 0, (size_t)kN * kD * sizeof(float), stream);

    edge_scores_kernel<<<edgeGrid, 256, 0, stream>>>(ei, q, kk, biasb, scores, menc);
    edge_soft_agg_kernel<<<edgeGrid, 256, 0, stream>>>(ei, scores, menc, vv, den, agg);
    normalize_agg_kernel<<<divUp(kN * kD, 256), 256, 0, stream>>>(agg, den);

    gemm128_wmma_kernel<<<gemmGrid, 128, 0, stream>>>(agg, Wo + wl, bo + bl, yo, kN);
    resid_ln_kernel<<<divUp(kN * 32, 256), 256, 0, stream>>>(yo, x, deg,
                                                             lng + bl, lnb + bl);
  }

  (void)hipMemsetAsync(pool, 0, kD * sizeof(float), stream);
  pool_reduce_kernel<<<256, 128, 0, stream>>>(x, pool);
  mlp_kernel<<<1, 128, 0, stream>>>(pool, pW1, pb1, pW2, pb2, emb);
}